// RDGATLayer_68590627717475
// MI455X (gfx1250) — compile-verified
//
#include <hip/hip_runtime.h>

typedef __attribute__((ext_vector_type(2))) float v2f;
typedef __attribute__((ext_vector_type(4))) float v4f;
typedef __attribute__((ext_vector_type(8))) float v8f;

#define F_DIM   128
#define MAX_REF 10
#define MAX_DIR 32
#define NDEPTH  4
#define LRELU_A 0.2f

__device__ __forceinline__ float wave_sum(float v) {
#pragma unroll
  for (int m = 16; m >= 1; m >>= 1) v += __shfl_xor(v, m, 32);
  return v;
}
__device__ __forceinline__ float wave_max(float v) {
#pragma unroll
  for (int m = 16; m >= 1; m >>= 1) v = fmaxf(v, __shfl_xor(v, m, 32));
  return v;
}
__device__ __forceinline__ float leaky(float x) { return x >= 0.f ? x : LRELU_A * x; }
__device__ __forceinline__ float sigmoidf(float x) { return 1.f / (1.f + __expf(-x)); }
__device__ __forceinline__ float dot4(v4f a, v4f b) {
  return a.x * b.x + a.y * b.y + a.z * b.z + a.w * b.w;
}

// ---------------------------------------------------------------------------
// Kernel 1: Wh_ref = h @ W_ref and Wh_dir = h @ W_dir via fp32 WMMA 16x16x4.
// One wave per 16-row strip; 8 N-tiles x (ref,dir) accumulators = 16 v8f.
// A frag: lanes 0-15 hold A[m][k],A[m][k+1]; lanes 16-31 hold A[m][k+2],A[m][k+3].
// B frag: lanes 0-15 hold B[k][n],B[k+1][n]; lanes 16-31 hold B[k+2][n],B[k+3][n].
// ---------------------------------------------------------------------------
__global__ __launch_bounds__(256) void wh_gemm(
    const float* __restrict__ h, const float* __restrict__ Wr,
    const float* __restrict__ Wd, float* __restrict__ WhR,
    float* __restrict__ WhD, int N) {
  const int lane  = threadIdx.x & 31;
  const int wave  = threadIdx.x >> 5;
  const int mtile = blockIdx.x * 8 + wave;
  if (mtile * 16 >= N) return;               // whole-wave exit: EXEC stays full
  const int row0  = mtile * 16;
  const int l16   = lane & 15;
  const int khalf = (lane >> 4) << 1;        // 0 or 2
  int arow_idx = row0 + l16;
  if (arow_idx >= N) arow_idx = N - 1;       // clamp (N % 16 == 0 here anyway)
  const float* arow = h + (size_t)arow_idx * F_DIM;

  v8f accR[8], accD[8];
  const v8f vz = {0.f, 0.f, 0.f, 0.f, 0.f, 0.f, 0.f, 0.f};
#pragma unroll
  for (int t = 0; t < 8; ++t) { accR[t] = vz; accD[t] = vz; }

#pragma unroll 4
  for (int kk = 0; kk < 32; ++kk) {
    const int k = kk * 4 + khalf;
    v2f a = *(const v2f*)(arow + k);
#pragma unroll
    for (int t = 0; t < 8; ++t) {
      const int nn = t * 16 + l16;
      v2f bR;
      bR.x = Wr[(size_t)k * F_DIM + nn];
      bR.y = Wr[(size_t)(k + 1) * F_DIM + nn];
      accR[t] = __builtin_amdgcn_wmma_f32_16x16x4_f32(
          false, a, false, bR, (short)0, accR[t], false, false);
      v2f bD;
      bD.x = Wd[(size_t)k * F_DIM + nn];
      bD.y = Wd[(size_t)(k + 1) * F_DIM + nn];
      accD[t] = __builtin_amdgcn_wmma_f32_16x16x4_f32(
          false, a, false, bD, (short)0, accD[t], false, false);
    }
  }

  // C/D layout: VGPR g -> row (row0 + g) for lanes 0-15, row (row0 + g + 8)
  // for lanes 16-31; column = t*16 + l16.
  const int rbase = row0 + ((lane >> 4) << 3);
#pragma unroll
  for (int g = 0; g < 8; ++g) {
    const int r = rbase + g;
    if (r >= N) continue;
    float* oR = WhR + (size_t)r * F_DIM + l16;
    float* oD = WhD + (size_t)r * F_DIM + l16;
#pragma unroll
    for (int t = 0; t < 8; ++t) {
      oR[t * 16] = accR[t][g];
      oD[t * 16] = accD[t][g];
    }
  }
}

// ---------------------------------------------------------------------------
// Kernel 2: ref-branch attention. One wave per node; each lane owns 4 features.
// Writes 0.5 * sigmoid(out_ref).
// ---------------------------------------------------------------------------
__global__ __launch_bounds__(256) void ref_attn(
    const float* __restrict__ Wh, const float* __restrict__ a_ref,
    const int* __restrict__ nbr, const int* __restrict__ cnts,
    float* __restrict__ out, int N) {
  const int lane = threadIdx.x & 31;
  const int n = blockIdx.x * 8 + (threadIdx.x >> 5);
  if (n >= N) return;
  const int fo = lane * 4;

  const v4f aself = *(const v4f*)(a_ref + fo);
  const v4f anbr  = *(const v4f*)(a_ref + F_DIM + fo);
  const v4f wh    = *(const v4f*)(Wh + (size_t)n * F_DIM + fo);
  const float s_self = wave_sum(dot4(wh, aself));

  v4f f[NDEPTH];
  float e[NDEPTH];
#pragma unroll
  for (int d = 0; d < NDEPTH; ++d) {
    const int cnt = cnts[(size_t)n * NDEPTH + d];
    const int* nb = nbr + ((size_t)n * NDEPTH + d) * MAX_REF;
    v4f acc = {0.f, 0.f, 0.f, 0.f};
    for (int k = 0; k < cnt; ++k) {
      const int idx = nb[k];
      acc += *(const v4f*)(Wh + (size_t)idx * F_DIM + fo);
    }
    const float inv = 1.f / (float)(cnt > 0 ? cnt : 1);
    acc = acc * inv;
    f[d] = acc;
    e[d] = leaky(s_self + wave_sum(dot4(acc, anbr)));
  }

  const float m = fmaxf(fmaxf(e[0], e[1]), fmaxf(e[2], e[3]));
  float p[NDEPTH], s = 0.f;
#pragma unroll
  for (int d = 0; d < NDEPTH; ++d) { p[d] = __expf(e[d] - m); s += p[d]; }
  const float invs = 1.f / s;
  v4f o = {0.f, 0.f, 0.f, 0.f};
#pragma unroll
  for (int d = 0; d < NDEPTH; ++d) o += f[d] * (p[d] * invs);

  float* ob = out + (size_t)n * F_DIM + fo;
  ob[0] = 0.5f * sigmoidf(o.x);
  ob[1] = 0.5f * sigmoidf(o.y);
  ob[2] = 0.5f * sigmoidf(o.z);
  ob[3] = 0.5f * sigmoidf(o.w);
}

// ---------------------------------------------------------------------------
// Kernel 3: dir-branch attention. One wave per node; lane k keeps e_k.
// Second gather pass (rows are L2-resident). Accumulates 0.5 * sigmoid(out_dir).
// ---------------------------------------------------------------------------
__global__ __launch_bounds__(256) void dir_attn(
    const float* __restrict__ Wh, const float* __restrict__ a_dir,
    const int* __restrict__ nbr, const int* __restrict__ cnts,
    float* __restrict__ out, int N) {
  const int lane = threadIdx.x & 31;
  const int n = blockIdx.x * 8 + (threadIdx.x >> 5);
  if (n >= N) return;
  const int fo = lane * 4;

  const v4f aself = *(const v4f*)(a_dir + fo);
  const v4f anbr  = *(const v4f*)(a_dir + F_DIM + fo);
  const v4f wh    = *(const v4f*)(Wh + (size_t)n * F_DIM + fo);
  const float s_self = wave_sum(dot4(wh, aself));

  const int cnt = cnts[n];                    // 1..32 per reference
  const int* nb = nbr + (size_t)n * MAX_DIR;

  float e_mine = -__builtin_inff();           // lane k holds e_k
  for (int k = 0; k < cnt; ++k) {
    const int idx = nb[k];
    const v4f r = *(const v4f*)(Wh + (size_t)idx * F_DIM + fo);
    const float ek = leaky(s_self + wave_sum(dot4(r, anbr)));
    if (lane == k) e_mine = ek;
  }

  const float m = wave_max(e_mine);
  const float p = (lane < cnt) ? __expf(e_mine - m) : 0.f;
  const float s = wave_sum(p);
  const float alpha_mine = p / s;

  v4f o = {0.f, 0.f, 0.f, 0.f};
  for (int k = 0; k < cnt; ++k) {
    const int idx = nb[k];
    const v4f r = *(const v4f*)(Wh + (size_t)idx * F_DIM + fo);
    const float a = __shfl(alpha_mine, k, 32);
    o += r * a;
  }

  float* ob = out + (size_t)n * F_DIM + fo;   // ref_attn ran first on this stream
  ob[0] += 0.5f * sigmoidf(o.x);
  ob[1] += 0.5f * sigmoidf(o.y);
  ob[2] += 0.5f * sigmoidf(o.z);
  ob[3] += 0.5f * sigmoidf(o.w);
}

extern "C" void kernel_launch(void* const* d_in, const int* in_sizes, int n_in,
                              void* d_out, int out_size, void* d_ws, size_t ws_size,
                              hipStream_t stream) {
  const float* h    = (const float*)d_in[0];
  const float* Wref = (const float*)d_in[1];
  const float* aref = (const float*)d_in[2];
  const float* Wdir = (const float*)d_in[3];
  const float* adir = (const float*)d_in[4];
  const int* rnbr   = (const int*)d_in[5];
  const int* rcnt   = (const int*)d_in[6];
  const int* dnbr   = (const int*)d_in[7];
  const int* dcnt   = (const int*)d_in[8];
  float* out = (float*)d_out;

  const int N = in_sizes[0] / F_DIM;

  float* WhR = (float*)d_ws;                       // N*128 floats
  float* WhD = WhR + (size_t)N * F_DIM;            // N*128 floats

  const int mtiles = (N + 15) / 16;
  wh_gemm<<<(mtiles + 7) / 8, 256, 0, stream>>>(h, Wref, Wdir, WhR, WhD, N);
  ref_attn<<<(N + 7) / 8, 256, 0, stream>>>(WhR, aref, rnbr, rcnt, out, N);
  dir_attn<<<(N + 7) / 8, 256, 0, stream>>>(WhD, adir, dnbr, dcnt, out, N);
}